// SimpleRuNNer_24103356465417
// MI455X (gfx1250) — compile-verified
//
#include <hip/hip_runtime.h>
#include <hip/hip_bf16.h>

// Problem constants (from reference)
#define S   2048
#define W   96
#define EE  100      // embedding dim
#define HH  100      // hidden dim
#define PD  50
#define HP  112      // H padded to 7 WMMA n-tiles of 16

typedef __attribute__((ext_vector_type(2))) float v2f;
typedef __attribute__((ext_vector_type(8))) float v8f;

// ---------------------------------------------------------------------------
// Kernel 1: ragged mean-pool of word embeddings -> sent (S x E)
// One block per sentence; 128 threads (4 waves).
// ---------------------------------------------------------------------------
__global__ void runner_pool_kernel(const int* __restrict__ x,
                                   const float* __restrict__ word_emb,
                                   float* __restrict__ sent) {
    const int s = blockIdx.x;
    __shared__ int len_sh;
    if (threadIdx.x == 0) len_sh = 0;
    __syncthreads();
    if (threadIdx.x < W) {
        if (x[s * W + threadIdx.x] != 0) atomicAdd(&len_sh, 1);
    }
    __syncthreads();
    const int L = len_sh;
    const float inv = 1.0f / (float)(L > 0 ? L : 1);
    const int e = threadIdx.x;
    if (e < EE) {
        float acc = 0.0f;
        for (int w = 0; w < L; ++w) {
            const int tok = x[s * W + w];
            acc += word_emb[tok * EE + e];
        }
        sent[s * EE + e] = acc * inv;
    }
}

// ---------------------------------------------------------------------------
// Kernel 2: doc_feat = mean_s sent ; doc = tanh(doc_feat @ fc1_w^T + fc1_b);
//           v = Wc + Ws @ doc   (folds (h@Ws)@doc == h@(Ws@doc))
// Single block of 128 threads.
// ---------------------------------------------------------------------------
__global__ void runner_docv_kernel(const float* __restrict__ sent,
                                   const float* __restrict__ fc1_w,
                                   const float* __restrict__ fc1_b,
                                   const float* __restrict__ Wc,
                                   const float* __restrict__ Ws,
                                   float* __restrict__ v) {
    __shared__ float df[EE];
    __shared__ float dd[HH];
    const int t = threadIdx.x;
    if (t < EE) {
        float acc = 0.0f;
        for (int s = 0; s < S; ++s) acc += sent[s * EE + t];
        df[t] = acc * (1.0f / (float)S);
    }
    __syncthreads();
    if (t < HH) {
        float acc = fc1_b[t];
        for (int e = 0; e < EE; ++e) acc += df[e] * fc1_w[t * EE + e];
        dd[t] = tanhf(acc);
    }
    __syncthreads();
    if (t < HH) {
        float acc = Wc[t];
        for (int j = 0; j < HH; ++j) acc += Ws[t * HH + j] * dd[j];
        v[t] = acc;
    }
}

// ---------------------------------------------------------------------------
// Kernel 3 (WMMA): h = tanh(sent @ fc2_w^T + fc2_b), stored S x HP with
// zero-padded columns [H, HP). One wave per 16x16 output tile.
// A(16x4 f32): lane<16 -> row lr, K=k0..k0+1 ; lane>=16 -> K=k0+2..k0+3.
// B(4x16 f32): lane<16 -> col lr, K=k0..k0+1 ; lane>=16 -> K=k0+2..k0+3.
// B[k][n] = fc2_w[n][k] -> contiguous float2 loads from fc2_w rows.
// ---------------------------------------------------------------------------
__global__ void runner_gemm_h_kernel(const float* __restrict__ sent,
                                     const float* __restrict__ fc2_w,
                                     const float* __restrict__ fc2_b,
                                     float* __restrict__ hbuf) {
    const int mt = blockIdx.x, nt = blockIdx.y;
    const int lane = threadIdx.x;
    const int half = lane >> 4;
    const int lr   = lane & 15;
    const int arow = mt * 16 + lr;
    const int ncol = nt * 16 + lr;
    const int nc   = (ncol < HH) ? ncol : (HH - 1);   // clamp edge-tile loads

    const float* aptr = sent + arow * EE + 2 * half;
    const float* bptr = fc2_w + nc * EE + 2 * half;

    v8f acc = {};
    #pragma unroll
    for (int k0 = 0; k0 < EE; k0 += 4) {
        v2f a = *(const v2f*)(aptr + k0);
        v2f b = *(const v2f*)(bptr + k0);
        acc = __builtin_amdgcn_wmma_f32_16x16x4_f32(
            false, a, false, b, (short)0, acc, false, false);
    }

    const bool valid = (ncol < HH);
    const float bias = valid ? fc2_b[ncol] : 0.0f;
    #pragma unroll
    for (int r = 0; r < 8; ++r) {
        const int row = mt * 16 + half * 8 + r;       // C/D vgpr layout
        const float val = valid ? tanhf(acc[r] + bias) : 0.0f;
        hbuf[row * HP + ncol] = val;
    }
}

// ---------------------------------------------------------------------------
// Kernel 4 (WMMA): hWr = h @ Wr, stored S x HP with zero-padded cols.
// B = Wr directly (row-major KxN); b components strided by H.
// ---------------------------------------------------------------------------
__global__ void runner_gemm_hwr_kernel(const float* __restrict__ hbuf,
                                       const float* __restrict__ Wr,
                                       float* __restrict__ hwr) {
    const int mt = blockIdx.x, nt = blockIdx.y;
    const int lane = threadIdx.x;
    const int half = lane >> 4;
    const int lr   = lane & 15;
    const int arow = mt * 16 + lr;
    const int ncol = nt * 16 + lr;
    const int nc   = (ncol < HH) ? ncol : (HH - 1);

    v8f acc = {};
    #pragma unroll
    for (int k0 = 0; k0 < HH; k0 += 4) {
        v2f a = *(const v2f*)(hbuf + arow * HP + k0 + 2 * half);
        const int kb = k0 + 2 * half;
        v2f b;
        b.x = Wr[kb * HH + nc];
        b.y = Wr[(kb + 1) * HH + nc];
        acc = __builtin_amdgcn_wmma_f32_16x16x4_f32(
            false, a, false, b, (short)0, acc, false, false);
    }

    const bool valid = (ncol < HH);
    #pragma unroll
    for (int r = 0; r < 8; ++r) {
        const int row = mt * 16 + half * 8 + r;
        hwr[row * HP + ncol] = valid ? acc[r] : 0.0f;
    }
}

// ---------------------------------------------------------------------------
// Kernel 5: pre[s] = b + h[s,:]·v + pos_emb[s,:]·Wp
// ---------------------------------------------------------------------------
__global__ void runner_pre_kernel(const float* __restrict__ hbuf,
                                  const float* __restrict__ v,
                                  const float* __restrict__ pos_emb,
                                  const float* __restrict__ Wp,
                                  const float* __restrict__ bsc,
                                  float* __restrict__ pre) {
    const int s = blockIdx.x * blockDim.x + threadIdx.x;
    if (s >= S) return;
    float acc = bsc[0];
    for (int i = 0; i < HH; ++i) acc += hbuf[s * HP + i] * v[i];
    for (int j = 0; j < PD; ++j) acc += pos_emb[s * PD + j] * Wp[j];
    pre[s] = acc;
}

// ---------------------------------------------------------------------------
// Kernel 6: sequential nonlinear scan (not parallelizable). Single wave32:
// lanes 0..27 each hold 4 state elements (HP=112 cols, pads are exact zeros).
// Per step: 4x v_tanh, float4 loads, shfl_xor butterfly reduce, sigmoid.
// ---------------------------------------------------------------------------
__global__ void runner_scan_kernel(const float* __restrict__ hbuf,
                                   const float* __restrict__ hwr,
                                   const float* __restrict__ pre,
                                   float* __restrict__ probs) {
    const int lane = threadIdx.x;
    const bool act = (lane * 4) < HP;                 // lanes 0..27
    float4 s4 = make_float4(0.f, 0.f, 0.f, 0.f);

    for (int t = 0; t < S; ++t) {
        float partial = 0.0f;
        float4 hw = make_float4(0.f, 0.f, 0.f, 0.f);
        float4 hv = make_float4(0.f, 0.f, 0.f, 0.f);
        if (act) {
            hw = *(const float4*)(hwr + t * HP + lane * 4);
            hv = *(const float4*)(hbuf + t * HP + lane * 4);
            if (t + 1 < S) {                          // global_prefetch_b8
                __builtin_prefetch(hwr + (t + 1) * HP + lane * 4, 0, 3);
                __builtin_prefetch(hbuf + (t + 1) * HP + lane * 4, 0, 3);
            }
            partial = hw.x * tanhf(s4.x) + hw.y * tanhf(s4.y)
                    + hw.z * tanhf(s4.z) + hw.w * tanhf(s4.w);
        }
        // wave32 butterfly reduction (all lanes participate)
        #pragma unroll
        for (int off = 16; off > 0; off >>= 1)
            partial += __shfl_xor(partial, off, 32);

        const float logit = pre[t] - partial;
        const float prob  = 1.0f / (1.0f + expf(-logit));
        if (lane == 0) probs[t] = prob;
        if (act) {
            s4.x += hv.x * prob; s4.y += hv.y * prob;
            s4.z += hv.z * prob; s4.w += hv.w * prob;
        }
    }
}

// ---------------------------------------------------------------------------
// Launch
// ---------------------------------------------------------------------------
extern "C" void kernel_launch(void* const* d_in, const int* in_sizes, int n_in,
                              void* d_out, int out_size, void* d_ws, size_t ws_size,
                              hipStream_t stream) {
    const int*   x        = (const int*)  d_in[0];
    const float* word_emb = (const float*)d_in[1];
    const float* pos_emb  = (const float*)d_in[2];
    const float* fc1_w    = (const float*)d_in[3];
    const float* fc1_b    = (const float*)d_in[4];
    const float* fc2_w    = (const float*)d_in[5];
    const float* fc2_b    = (const float*)d_in[6];
    const float* Wc       = (const float*)d_in[7];
    const float* Ws       = (const float*)d_in[8];
    const float* Wr       = (const float*)d_in[9];
    const float* Wp       = (const float*)d_in[10];
    const float* bsc      = (const float*)d_in[11];
    float* probs = (float*)d_out;

    // Workspace layout (floats)
    float* ws   = (float*)d_ws;
    float* sent = ws;                       // S*E   = 204800
    float* vvec = sent + S * EE;            // 128
    float* hbuf = vvec + 128;               // S*HP  = 229376
    float* hwr  = hbuf + S * HP;            // S*HP  = 229376
    float* pre  = hwr + S * HP;             // S     = 2048

    runner_pool_kernel<<<S, 128, 0, stream>>>(x, word_emb, sent);
    runner_docv_kernel<<<1, 128, 0, stream>>>(sent, fc1_w, fc1_b, Wc, Ws, vvec);
    runner_gemm_h_kernel<<<dim3(S / 16, HP / 16), 32, 0, stream>>>(sent, fc2_w, fc2_b, hbuf);
    runner_gemm_hwr_kernel<<<dim3(S / 16, HP / 16), 32, 0, stream>>>(hbuf, Wr, hwr);
    runner_pre_kernel<<<(S + 255) / 256, 256, 0, stream>>>(hbuf, vvec, pos_emb, Wp, bsc, pre);
    runner_scan_kernel<<<1, 32, 0, stream>>>(hbuf, hwr, pre, probs);
}